// HierarchicalGraphPooling_14396730376425
// MI455X (gfx1250) — compile-verified
//
// HierarchicalGraphPooling forward for MI455X (gfx1250, CDNA5, wave32).
//
// Dense matmuls use V_WMMA_F32_16X16X4_F32 (exact fp32, matches reference).
// GEMM tiles are staged memory->LDS with the CDNA5 async copy engine
// (global_load_async_to_lds_b128, ASYNCcnt, s_wait_asynccnt) and double-
// buffered LDS so the copy of chunk t+1 overlaps the WMMAs of chunk t.
// Edge aggregation uses f32 global atomics (whole working set < 128MB, lives
// in the 192MB L2). Per-graph top-k is an in-LDS bitonic sort (npg is a
// power of two, graphs are contiguous equal-size blocks).
//
// d_in order (JAX pytree flatten: dict keys sorted alphabetically):
//  0 x[32768,640] 1 edge_index[2,524288] 2 batch[32768] 3 features[16,66]
//  4 feat.b 5 feat.bn 6 feat.g 7 feat.w[66,256]
//  8..25 gcn[i][j].{b,w} i=0..2 j=0..2  (b then w)
//  26 head.b1 27 head.b2 28 head.b3 29 head.bn1 30 head.g1
//  31 head.w1[1280,256] 32 head.w2[256,128] 33 head.w3[128]
//  34 input_proj.b 35 input_proj.w[640,256]
//  36 ln0.b 37 ln0.g 38 ln1.b 39 ln1.g 40 ln2.b 41 ln2.g
//  42 pool0 43 pool1 44 pool2
#include <hip/hip_runtime.h>
#include <math.h>

typedef __attribute__((ext_vector_type(2))) float v2f;
typedef __attribute__((ext_vector_type(8))) float v8f;

#define HD    256
#define NB    16
#define NEDGE 524288
#define NT0   32768

__device__ __forceinline__ float gelu_exact(float x) {
  return 0.5f * x * (1.0f + erff(x * 0.70710678118654752f));
}

// ---- CDNA5 async memory->LDS copy (ASYNCcnt-tracked, no VGPR data) --------
// LDS operand is the 32-bit LDS byte offset (generic LDS addr low 32 bits).
__device__ __forceinline__ void async_g2l_b128(unsigned lds_off,
                                               const float* gptr) {
  unsigned long long ga = (unsigned long long)(uintptr_t)gptr;
  asm volatile("global_load_async_to_lds_b128 %0, %1, off"
               :: "v"(lds_off), "v"(ga)
               : "memory");
}
__device__ __forceinline__ void wait_async_le3() {
  asm volatile("s_wait_asynccnt 0x3" ::: "memory");
}
__device__ __forceinline__ void wait_async_le0() {
  asm volatile("s_wait_asynccnt 0x0" ::: "memory");
}
__device__ __forceinline__ unsigned lds_off_of(const void* p) {
  return (unsigned)(uintptr_t)p;
}

// ---------------------------------------------------------------------------
// Tiled fp32 WMMA GEMM: C[M,N] = A[M,K] @ W[K,N] (+bias, +gelu).
// Block 256 thr = 8 waves as 4x2 grid of 16x16 WMMA tiles -> 64x32 block tile.
// Double-buffered LDS chunks (row stride 36 floats: 16B-aligned b128 async
// stores, 8B-aligned v2f A-fragment reads), filled by the async copy engine.
// ---------------------------------------------------------------------------
__global__ __launch_bounds__(256) void k_gemm_wmma(
    const float* __restrict__ A, const float* __restrict__ W,
    float* __restrict__ C, const float* __restrict__ bias,
    int M, int N, int K, int fuse_gelu) {
  __shared__ float As[2][64 * 36];
  __shared__ float Bs[2][32 * 36];
  const int tid  = threadIdx.x;
  const int lane = tid & 31;
  const int wave = tid >> 5;
  const int wrow = wave >> 1;   // 0..3
  const int wcol = wave & 1;    // 0..1
  const int half = lane >> 4;
  const int l15  = lane & 15;
  const int rowBase = blockIdx.x * 64;
  const int colBase = blockIdx.y * 32;

  v8f acc = {0.f, 0.f, 0.f, 0.f, 0.f, 0.f, 0.f, 0.f};
  const int amr = wrow * 16 + l15;  // LDS A row for this lane's fragment
  const int bnc = wcol * 16 + l15;  // LDS B col for this lane's fragment

  // Per-thread cooperative-copy slots: A tile 64x32 = 512 float4 (2/thread),
  // B tile 32x32 = 256 float4 (1/thread).
  const int rA = tid >> 3;            // 0..31
  const int c4 = (tid & 7) << 2;      // 0,4,..,28
  const float* gA0 = &A[(size_t)(rowBase + rA) * K + c4];
  const float* gA1 = &A[(size_t)(rowBase + rA + 32) * K + c4];
  const float* gB  = &W[(size_t)rA * N + colBase + c4];
  unsigned lA0[2], lA1[2], lB[2];
#pragma unroll
  for (int b = 0; b < 2; ++b) {
    lA0[b] = lds_off_of(&As[b][rA * 36 + c4]);
    lA1[b] = lds_off_of(&As[b][(rA + 32) * 36 + c4]);
    lB[b]  = lds_off_of(&Bs[b][rA * 36 + c4]);
  }

  const int nT = K >> 5;  // K is a multiple of 32
  // prologue: async-fill buffer 0 with chunk 0
  async_g2l_b128(lA0[0], gA0);
  async_g2l_b128(lA1[0], gA1);
  async_g2l_b128(lB[0],  gB);

  for (int t = 0; t < nT; ++t) {
    const int cur = t & 1;
    if (t + 1 < nT) {  // overlap: issue chunk t+1 into the other buffer
      const int nxt = cur ^ 1;
      const size_t ka = (size_t)(t + 1) * 32;        // A col advance
      const size_t kb = (size_t)(t + 1) * 32 * N;    // B row advance
      async_g2l_b128(lA0[nxt], gA0 + ka);
      async_g2l_b128(lA1[nxt], gA1 + ka);
      async_g2l_b128(lB[nxt],  gB + kb);
      wait_async_le3();   // in-order: chunk t landed, chunk t+1 in flight
    } else {
      wait_async_le0();
    }
    __syncthreads();
    const float* as = As[cur];
    const float* bs = Bs[cur];
#pragma unroll
    for (int kk = 0; kk < 32; kk += 4) {
      // A frag (16x4 f32): v0=A[m][k0+2h], v1=A[m][k0+2h+1] (contiguous pair)
      v2f af = *(const v2f*)&as[amr * 36 + kk + 2 * half];
      // B frag (4x16 f32): v0=B[k0+2h][n], v1=B[k0+2h+1][n]
      v2f bf;
      bf.x = bs[(kk + 2 * half) * 36 + bnc];
      bf.y = bs[(kk + 2 * half + 1) * 36 + bnc];
      acc = __builtin_amdgcn_wmma_f32_16x16x4_f32(false, af, false, bf,
                                                  (short)0, acc, false, false);
    }
    if (t + 1 < nT) __syncthreads();  // buffer `cur` is refilled at t+2
  }
  // C/D layout: vgpr r -> M=r (lanes 0-15) / M=r+8 (lanes 16-31), N=lane&15
  const int nglob = colBase + wcol * 16 + l15;
#pragma unroll
  for (int r = 0; r < 8; ++r) {
    int m = rowBase + wrow * 16 + r + 8 * half;
    float v = acc[r];
    if (bias) v += bias[nglob];
    if (fuse_gelu) v = gelu_exact(v);
    C[(size_t)m * N + nglob] = v;
  }
}

// ---------------------------------------------------------------------------
// Head GEMM: C[16,N] = A[16,K] @ W[K,N] + bias. One block, 16 waves, each a
// 16x16 WMMA tile over the full K. Operands direct from L2 (tiny).
// ---------------------------------------------------------------------------
__global__ __launch_bounds__(512) void k_head1_wmma(
    const float* __restrict__ A, const float* __restrict__ W,
    const float* __restrict__ bias, float* __restrict__ C, int K, int N) {
  const int lane = threadIdx.x & 31;
  const int wave = threadIdx.x >> 5;  // col tile 0..15
  const int half = lane >> 4;
  const int l15  = lane & 15;
  const int nb   = wave * 16;
  v8f acc = {0.f, 0.f, 0.f, 0.f, 0.f, 0.f, 0.f, 0.f};
  for (int kb = 0; kb < K; kb += 4) {
    v2f af = *(const v2f*)&A[(size_t)l15 * K + kb + 2 * half];
    v2f bf;
    bf.x = W[(size_t)(kb + 2 * half) * N + nb + l15];
    bf.y = W[(size_t)(kb + 2 * half + 1) * N + nb + l15];
    acc = __builtin_amdgcn_wmma_f32_16x16x4_f32(false, af, false, bf,
                                                (short)0, acc, false, false);
  }
  const int n = nb + l15;
#pragma unroll
  for (int r = 0; r < 8; ++r) {
    int m = r + 8 * half;
    C[(size_t)m * N + n] = acc[r] + bias[n];
  }
}

// ---------------------------------------------------------------------------
// LayerNorm over HD=256 features, one row per block, optional fused gelu.
// ---------------------------------------------------------------------------
__global__ __launch_bounds__(256) void k_layernorm(
    const float* __restrict__ X, float* __restrict__ Y,
    const float* __restrict__ g, const float* __restrict__ b, int fuse_gelu) {
  __shared__ float red[256];
  int row = blockIdx.x, f = threadIdx.x;
  float v = X[(size_t)row * HD + f];
  red[f] = v;
  __syncthreads();
  for (int s = 128; s > 0; s >>= 1) { if (f < s) red[f] += red[f + s]; __syncthreads(); }
  float mu = red[0] * (1.0f / HD);
  __syncthreads();
  float d = v - mu;
  red[f] = d * d;
  __syncthreads();
  for (int s = 128; s > 0; s >>= 1) { if (f < s) red[f] += red[f + s]; __syncthreads(); }
  float var = red[0] * (1.0f / HD);
  float y = d * rsqrtf(var + 1e-5f) * g[f] + b[f];
  if (fuse_gelu) y = gelu_exact(y);
  Y[(size_t)row * HD + f] = y;
}

// score[n] = tanh((x[n].p)/||p||), one node per block
__global__ __launch_bounds__(256) void k_score(
    const float* __restrict__ X, const float* __restrict__ p,
    float* __restrict__ score) {
  __shared__ float rd[256];
  __shared__ float rp[256];
  int n = blockIdx.x, f = threadIdx.x;
  float pf = p[f];
  rd[f] = X[(size_t)n * HD + f] * pf;
  rp[f] = pf * pf;
  __syncthreads();
  for (int s = 128; s > 0; s >>= 1) {
    if (f < s) { rd[f] += rd[f + s]; rp[f] += rp[f + s]; }
    __syncthreads();
  }
  if (f == 0) score[n] = tanhf(rd[0] * rsqrtf(rp[0]));
}

// mean-pool contiguous per-graph blocks into combined[:, slot*256 : +256]
__global__ __launch_bounds__(256) void k_meanpool(
    const float* __restrict__ X, float* __restrict__ combined, int npg, int slot) {
  int g = blockIdx.x, f = threadIdx.x;
  const float* xp = X + (size_t)g * npg * HD + f;
  float s = 0.f;
  for (int r = 0; r < npg; ++r) s += xp[(size_t)r * HD];
  combined[(size_t)g * 1280 + slot * HD + f] = s / (float)npg;
}

// per-graph descending bitonic sort of (score,idx); emit top-k perm + scores
__global__ __launch_bounds__(1024) void k_topk(
    const float* __restrict__ score, int npg, int k,
    int* __restrict__ perm, float* __restrict__ ssel) {
  __shared__ float s[2048];
  __shared__ int   id[2048];
  int g = blockIdx.x, t = threadIdx.x;
  for (int i = t; i < npg; i += 1024) { s[i] = score[g * npg + i]; id[i] = i; }
  __syncthreads();
  for (int sz = 2; sz <= npg; sz <<= 1) {
    for (int st = sz >> 1; st > 0; st >>= 1) {
      for (int i = t; i < npg; i += 1024) {
        int j = i ^ st;
        if (j > i) {
          bool desc = ((i & sz) == 0);
          bool sw = desc ? (s[i] < s[j]) : (s[i] > s[j]);
          if (sw) {
            float tv = s[i]; s[i] = s[j]; s[j] = tv;
            int ti = id[i]; id[i] = id[j]; id[j] = ti;
          }
        }
      }
      __syncthreads();
    }
  }
  for (int j = t; j < k; j += 1024) {
    perm[g * k + j] = g * npg + id[j];
    ssel[g * k + j] = s[j];
  }
}

__global__ void k_fill_f32(float* p, float v, int n) {
  int i = blockIdx.x * blockDim.x + threadIdx.x;
  if (i < n) p[i] = v;
}
__global__ void k_fill_i32(int* p, int v, int n) {
  int i = blockIdx.x * blockDim.x + threadIdx.x;
  if (i < n) p[i] = v;
}
__global__ void k_edges_init(const int* __restrict__ ei, int* __restrict__ row,
                             int* __restrict__ col, int* __restrict__ em) {
  int e = blockIdx.x * blockDim.x + threadIdx.x;
  if (e < NEDGE) { row[e] = ei[e]; col[e] = ei[NEDGE + e]; em[e] = 1; }
}
__global__ void k_deg(const int* __restrict__ col, const int* __restrict__ em,
                      float* __restrict__ deg) {
  int e = blockIdx.x * blockDim.x + threadIdx.x;
  if (e < NEDGE && em[e]) atomicAdd(&deg[col[e]], 1.0f);
}
__global__ void k_dis(const float* __restrict__ deg, float* __restrict__ dis, int n) {
  int i = blockIdx.x * blockDim.x + threadIdx.x;
  if (i < n) dis[i] = rsqrtf(deg[i] + 1.0f);
}

// agg[col] += xw[row] * dis[row]*dis[col]; lane layout: 32 f-chunks x 8 edges
__global__ __launch_bounds__(256) void k_scatter(
    const float* __restrict__ xw, const int* __restrict__ row,
    const int* __restrict__ col, const int* __restrict__ em,
    const float* __restrict__ dis, float* __restrict__ agg) {
  int t = blockIdx.x * 256 + threadIdx.x;
  int e = t >> 5;
  int f0 = (t & 31) << 3;
  if (e >= NEDGE || !em[e]) return;
  int r = row[e], c = col[e];
  float coeff = dis[r] * dis[c];
  const float4* xv = (const float4*)&xw[(size_t)r * HD + f0];
  float4 a = xv[0], b = xv[1];
  float* ap = &agg[(size_t)c * HD + f0];
  atomicAdd(ap + 0, a.x * coeff); atomicAdd(ap + 1, a.y * coeff);
  atomicAdd(ap + 2, a.z * coeff); atomicAdd(ap + 3, a.w * coeff);
  atomicAdd(ap + 4, b.x * coeff); atomicAdd(ap + 5, b.y * coeff);
  atomicAdd(ap + 6, b.z * coeff); atomicAdd(ap + 7, b.w * coeff);
}

// x = gelu(agg + xw*dis^2 + bias), 4 features per thread
__global__ void k_gcn_epilogue(float* __restrict__ x, const float* __restrict__ agg,
                               const float* __restrict__ xw, const float* __restrict__ dis,
                               const float* __restrict__ bias, int n) {
  int t = blockIdx.x * blockDim.x + threadIdx.x;
  if (t >= n * 64) return;
  int node = t >> 6;
  int f4 = (t & 63) << 2;
  float dd = dis[node]; dd *= dd;
  float4 ag = *(const float4*)&agg[(size_t)node * HD + f4];
  float4 xv = *(const float4*)&xw[(size_t)node * HD + f4];
  float4 o;
  o.x = gelu_exact(ag.x + xv.x * dd + bias[f4 + 0]);
  o.y = gelu_exact(ag.y + xv.y * dd + bias[f4 + 1]);
  o.z = gelu_exact(ag.z + xv.z * dd + bias[f4 + 2]);
  o.w = gelu_exact(ag.w + xv.w * dd + bias[f4 + 3]);
  *(float4*)&x[(size_t)node * HD + f4] = o;
}

__global__ void k_newid_scatter(int* __restrict__ newid, const int* __restrict__ perm,
                                int nnew) {
  int j = blockIdx.x * blockDim.x + threadIdx.x;
  if (j < nnew) newid[perm[j]] = j;
}
__global__ void k_gather(const float* __restrict__ xo, float* __restrict__ xn,
                         const int* __restrict__ perm, const float* __restrict__ ssel,
                         int nnew) {
  int t = blockIdx.x * blockDim.x + threadIdx.x;
  if (t >= nnew * 64) return;
  int j = t >> 6;
  int f4 = (t & 63) << 2;
  int src = perm[j];
  float sc = ssel[j];
  float4 v = *(const float4*)&xo[(size_t)src * HD + f4];
  v.x *= sc; v.y *= sc; v.z *= sc; v.w *= sc;
  *(float4*)&xn[(size_t)j * HD + f4] = v;
}
__global__ void k_reindex(int* __restrict__ row, int* __restrict__ col,
                          int* __restrict__ em, const int* __restrict__ newid) {
  int e = blockIdx.x * blockDim.x + threadIdx.x;
  if (e >= NEDGE) return;
  int m = em[e];
  int r2 = newid[row[e]];
  int c2 = newid[col[e]];
  int valid = (m && r2 >= 0 && c2 >= 0) ? 1 : 0;
  row[e] = valid ? r2 : 0;
  col[e] = valid ? c2 : 0;
  em[e] = valid;
}

// f = gelu(LN(features @ Wf + bf)) -> combined[:, 1024:1280]
__global__ __launch_bounds__(256) void k_feat(
    const float* __restrict__ feats, const float* __restrict__ W,
    const float* __restrict__ bias, const float* __restrict__ g,
    const float* __restrict__ b, float* __restrict__ combined) {
  __shared__ float red[256];
  int gr = blockIdx.x, f = threadIdx.x;
  float s = bias[f];
  for (int k = 0; k < 66; ++k) s += feats[gr * 66 + k] * W[k * HD + f];
  red[f] = s;
  __syncthreads();
  for (int st = 128; st > 0; st >>= 1) { if (f < st) red[f] += red[f + st]; __syncthreads(); }
  float mu = red[0] * (1.0f / HD);
  __syncthreads();
  float d = s - mu;
  red[f] = d * d;
  __syncthreads();
  for (int st = 128; st > 0; st >>= 1) { if (f < st) red[f] += red[f + st]; __syncthreads(); }
  float var = red[0] * (1.0f / HD);
  float y = d * rsqrtf(var + 1e-5f) * g[f] + b[f];
  combined[(size_t)gr * 1280 + 1024 + f] = gelu_exact(y);
}

__global__ __launch_bounds__(128) void k_head2(
    const float* __restrict__ y1, const float* __restrict__ W,
    const float* __restrict__ b, float* __restrict__ y2) {
  int g = blockIdx.x, j = threadIdx.x;
  float s = b[j];
  for (int k = 0; k < HD; ++k) s += y1[g * HD + k] * W[k * 128 + j];
  y2[g * 128 + j] = gelu_exact(s);
}
__global__ __launch_bounds__(128) void k_head3(
    const float* __restrict__ y2, const float* __restrict__ W,
    const float* __restrict__ b3, float* __restrict__ out) {
  __shared__ float red[128];
  int g = blockIdx.x, t = threadIdx.x;
  red[t] = y2[g * 128 + t] * W[t];
  __syncthreads();
  for (int s = 64; s > 0; s >>= 1) { if (t < s) red[t] += red[t + s]; __syncthreads(); }
  if (t == 0) out[g] = red[0] + b3[0];
}

extern "C" void kernel_launch(void* const* d_in, const int* in_sizes, int n_in,
                              void* d_out, int out_size, void* d_ws, size_t ws_size,
                              hipStream_t stream) {
  (void)in_sizes; (void)n_in; (void)out_size; (void)ws_size;
  const float* X     = (const float*)d_in[0];
  const int*   EI    = (const int*)d_in[1];
  const float* FEATS = (const float*)d_in[3];
  const float* featB  = (const float*)d_in[4];
  const float* featBN = (const float*)d_in[5];
  const float* featG  = (const float*)d_in[6];
  const float* featW  = (const float*)d_in[7];
  const float* gcnB[3][3]; const float* gcnW[3][3];
  for (int i = 0; i < 3; ++i)
    for (int j = 0; j < 3; ++j) {
      gcnB[i][j] = (const float*)d_in[8 + (i * 3 + j) * 2];
      gcnW[i][j] = (const float*)d_in[9 + (i * 3 + j) * 2];
    }
  const float* hB1  = (const float*)d_in[26];
  const float* hB2  = (const float*)d_in[27];
  const float* hB3  = (const float*)d_in[28];
  const float* hBN1 = (const float*)d_in[29];
  const float* hG1  = (const float*)d_in[30];
  const float* hW1  = (const float*)d_in[31];
  const float* hW2  = (const float*)d_in[32];
  const float* hW3  = (const float*)d_in[33];
  const float* ipB  = (const float*)d_in[34];
  const float* ipW  = (const float*)d_in[35];
  const float* lnB[3] = {(const float*)d_in[36], (const float*)d_in[38], (const float*)d_in[40]};
  const float* lnG[3] = {(const float*)d_in[37], (const float*)d_in[39], (const float*)d_in[41]};
  const float* poolP[3] = {(const float*)d_in[42], (const float*)d_in[43], (const float*)d_in[44]};

  // workspace carve-up (256B aligned)
  size_t off = 0;
  char* base = (char*)d_ws;
  auto carve = [&](size_t bytes) -> char* {
    char* p = base + off;
    off += (bytes + 255) & ~(size_t)255;
    return p;
  };
  float* buf0  = (float*)carve((size_t)NT0 * HD * 4);
  float* buf1  = (float*)carve((size_t)NT0 * HD * 4);
  float* xw    = (float*)carve((size_t)NT0 * HD * 4);
  float* deg   = (float*)carve(NT0 * 4);
  float* dis   = (float*)carve(NT0 * 4);
  float* score = (float*)carve(NT0 * 4);
  float* ssel  = (float*)carve((NT0 / 2) * 4);
  float* comb  = (float*)carve(NB * 1280 * 4);
  float* y1    = (float*)carve(NB * HD * 4);
  float* y1g   = (float*)carve(NB * HD * 4);
  float* y2    = (float*)carve(NB * 128 * 4);
  int* rowb    = (int*)carve(NEDGE * 4);
  int* colb    = (int*)carve(NEDGE * 4);
  int* emb     = (int*)carve(NEDGE * 4);
  int* newid   = (int*)carve(NT0 * 4);
  int* perm    = (int*)carve((NT0 / 2) * 4);

  // 1) input projection + gelu: buf0 = gelu(X @ ipW + ipB)
  k_gemm_wmma<<<dim3(NT0 / 64, HD / 32), 256, 0, stream>>>(X, ipW, buf0, ipB,
                                                           NT0, HD, 640, 1);
  // repr 0
  k_meanpool<<<NB, 256, 0, stream>>>(buf0, comb, 2048, 0);
  // edge buffers
  k_edges_init<<<NEDGE / 256, 256, 0, stream>>>(EI, rowb, colb, emb);

  float* bufs[2] = {buf0, buf1};
  int p = 0;
  int npg = 2048;
  for (int lvl = 0; lvl < 3; ++lvl) {
    int n = NB * npg;
    int kk = npg / 2;
    float* x = bufs[p];
    float* agg = bufs[1 - p];  // pong buffer doubles as agg scratch
    // symmetric-norm degrees (w_edge = emask)
    k_fill_f32<<<(n + 255) / 256, 256, 0, stream>>>(deg, 0.f, n);
    k_deg<<<NEDGE / 256, 256, 0, stream>>>(colb, emb, deg);
    k_dis<<<(n + 255) / 256, 256, 0, stream>>>(deg, dis, n);
    for (int j = 0; j < 3; ++j) {
      k_gemm_wmma<<<dim3(n / 64, HD / 32), 256, 0, stream>>>(x, gcnW[lvl][j], xw,
                                                             nullptr, n, HD, HD, 0);
      k_fill_f32<<<(n * HD + 255) / 256, 256, 0, stream>>>(agg, 0.f, n * HD);
      k_scatter<<<NEDGE / 8, 256, 0, stream>>>(xw, rowb, colb, emb, dis, agg);
      k_gcn_epilogue<<<(n * 64) / 256, 256, 0, stream>>>(x, agg, xw, dis,
                                                         gcnB[lvl][j], n);
    }
    k_layernorm<<<n, 256, 0, stream>>>(x, x, lnG[lvl], lnB[lvl], 0);
    // top-k pooling
    k_score<<<n, 256, 0, stream>>>(x, poolP[lvl], score);
    k_topk<<<NB, 1024, 0, stream>>>(score, npg, kk, perm, ssel);
    int nnew = NB * kk;
    k_fill_i32<<<(n + 255) / 256, 256, 0, stream>>>(newid, -1, n);
    k_newid_scatter<<<(nnew + 255) / 256, 256, 0, stream>>>(newid, perm, nnew);
    k_gather<<<(nnew * 64) / 256, 256, 0, stream>>>(x, bufs[1 - p], perm, ssel, nnew);
    k_reindex<<<NEDGE / 256, 256, 0, stream>>>(rowb, colb, emb, newid);
    p ^= 1;
    k_meanpool<<<NB, 256, 0, stream>>>(bufs[p], comb, kk, lvl + 1);
    npg = kk;
  }

  // feature branch -> combined[:,1024:1280]
  k_feat<<<NB, 256, 0, stream>>>(FEATS, featW, featB, featG, featBN, comb);
  // head
  k_head1_wmma<<<1, 512, 0, stream>>>(comb, hW1, hB1, y1, 1280, HD);
  k_layernorm<<<NB, 256, 0, stream>>>(y1, y1g, hG1, hBN1, 1);
  k_head2<<<NB, 128, 0, stream>>>(y1g, hW2, hB2, y2);
  k_head3<<<NB, 128, 0, stream>>>(y2, hW3, hB3, (float*)d_out);
}